// TopKGate_83554293776537
// MI455X (gfx1250) — compile-verified
//
#include <hip/hip_runtime.h>
#include <hip/hip_bf16.h>
#include <math.h>

// CDNA5 wave32 f32 WMMA fragments
typedef float v2f __attribute__((ext_vector_type(2)));
typedef float v8f __attribute__((ext_vector_type(8)));

#define S_TOK   8192
#define DMODEL  1024
#define NEXP    64
#define CAP     128
#define BLK_TOK 128                 // tokens per workgroup (8 waves x 16)
#define NBLK    (S_TOK / BLK_TOK)   // 64

// workspace layout (byte offsets)
#define WS_IDX   0           // int[8192]   chosen expert per token
#define WS_GMX   (32*1024)   // float[8192] max gate per token
#define WS_HIST  (64*1024)   // int[64][64] per-block expert histogram
#define WS_GSUM  (80*1024)   // float[64][64] per-block per-expert gate sums
#define WS_SCAL  (96*1024)   // float[64][4] per-block scalar partials
#define WS_OFFS  (100*1024)  // int[64][64] exclusive per-expert block offsets

// ---------------------------------------------------------------------------
// Kernel 1: zero the entire output (combine_weights / dispatch_mask are ~all 0)
// ---------------------------------------------------------------------------
__global__ void k_zero(float4* __restrict__ out4, long n4,
                       float* __restrict__ out, long n) {
  long i = (long)blockIdx.x * blockDim.x + threadIdx.x;
  long stride = (long)gridDim.x * blockDim.x;
  float4 z = make_float4(0.f, 0.f, 0.f, 0.f);
  for (long p = i; p < n4; p += stride) out4[p] = z;
  long tail = n4 * 4 + i;
  if (tail < n) out[tail] = 0.f;
}

// ---------------------------------------------------------------------------
// Kernel 2: logits via V_WMMA_F32_16X16X4_F32 + softmax/argmax/stat partials
// ---------------------------------------------------------------------------
__global__ __launch_bounds__(256) void k_gate(const float* __restrict__ x,
                                              const float* __restrict__ W,
                                              int*   __restrict__ idx,
                                              float* __restrict__ gmx,
                                              int*   __restrict__ hist,
                                              float* __restrict__ gsumB,
                                              float* __restrict__ scalB) {
  __shared__ float logitsL[BLK_TOK][NEXP];  // 32 KB
  __shared__ float gsumL[NEXP];
  __shared__ int   histL[NEXP];
  __shared__ float scalL[4];

  const int tid = threadIdx.x;
  if (tid < NEXP) { gsumL[tid] = 0.f; histL[tid] = 0; }
  if (tid < 4)    { scalL[tid] = 0.f; }

  const int wave = tid >> 5;
  const int lane = tid & 31;
  const int m    = lane & 15;           // M (token within 16) / N (expert within 16)
  const int kOff = (lane >> 4) << 1;    // half-wave K split: 0 or 2
  const int row  = blockIdx.x * BLK_TOK + wave * 16 + m;

  // 16x64 logits tile per wave: 4 N-tiles of 16x16 f32 accumulators
  v8f acc0 = {}, acc1 = {}, acc2 = {}, acc3 = {};
  const float* __restrict__ xr = x + (size_t)row * DMODEL;

  #pragma unroll 2
  for (int k = 0; k < DMODEL; k += 4) {
    // A fragment (16x4): lane<16 holds K={k,k+1}, lane>=16 holds K={k+2,k+3}
    v2f a;
    a.x = xr[k + kOff];
    a.y = xr[k + kOff + 1];
    // B fragments (4x16 each): lane<16 -> K={k,k+1}, lane>=16 -> K={k+2,k+3}; N = lane&15
    const float* __restrict__ wb = W + (size_t)(k + kOff) * NEXP + m;
    v2f b0, b1, b2, b3;
    b0.x = wb[0];  b0.y = wb[NEXP + 0];
    b1.x = wb[16]; b1.y = wb[NEXP + 16];
    b2.x = wb[32]; b2.y = wb[NEXP + 32];
    b3.x = wb[48]; b3.y = wb[NEXP + 48];
    acc0 = __builtin_amdgcn_wmma_f32_16x16x4_f32(false, a, false, b0, (short)0, acc0, false, false);
    acc1 = __builtin_amdgcn_wmma_f32_16x16x4_f32(false, a, false, b1, (short)0, acc1, false, false);
    acc2 = __builtin_amdgcn_wmma_f32_16x16x4_f32(false, a, false, b2, (short)0, acc2, false, false);
    acc3 = __builtin_amdgcn_wmma_f32_16x16x4_f32(false, a, false, b3, (short)0, acc3, false, false);
  }

  // C/D layout: lane -> N=lane&15, VGPR j -> M=j+8*(lane>>4)
  const int nIdx  = lane & 15;
  const int mHalf = (lane >> 4) * 8;
  #pragma unroll
  for (int j = 0; j < 8; ++j) {
    const int r = wave * 16 + mHalf + j;
    logitsL[r][ 0 + nIdx] = acc0[j];
    logitsL[r][16 + nIdx] = acc1[j];
    logitsL[r][32 + nIdx] = acc2[j];
    logitsL[r][48 + nIdx] = acc3[j];
  }
  __syncthreads();

  if (tid < BLK_TOK) {
    const int r = tid;
    const int s = blockIdx.x * BLK_TOK + r;

    // pass 1: max + first-argmax (matches jnp.argmax tie-breaking)
    float mval = logitsL[r][0];
    int   am   = 0;
    for (int e = 1; e < NEXP; ++e) {
      float l = logitsL[r][e];
      if (l > mval) { mval = l; am = e; }
    }
    // pass 2: exp(l - m), sum
    float sumExp = 0.f;
    for (int e = 0; e < NEXP; ++e) {
      float ev = __expf(logitsL[r][e] - mval);
      logitsL[r][e] = ev;
      sumExp += ev;
    }
    const float inv = 1.0f / sumExp;            // == max gate (exp(m-lse))
    const float lse = mval + __logf(sumExp);
    // pass 3: gates + per-row stats + per-expert gate sums (staggered to
    // spread LDS atomic traffic over all 64 addresses each step)
    float sumg = 0.f, sumsq = 0.f, ent = 0.f;
    for (int i = 0; i < NEXP; ++i) {
      const int e = (r + i) & (NEXP - 1);
      const float g = logitsL[r][e] * inv;
      sumg  += g;
      sumsq += g * g;
      ent   -= g * __logf(g + 1e-9f);
      atomicAdd(&gsumL[e], g);
    }
    idx[s] = am;
    gmx[s] = inv;
    atomicAdd(&histL[am], 1);
    atomicAdd(&scalL[0], lse * lse);                       // -> l_z
    atomicAdd(&scalL[1], ent);                             // -> batch_entropy
    atomicAdd(&scalL[2], inv);                             // -> batch_prob
    atomicAdd(&scalL[3], sumg / (sqrtf(sumsq) + 1e-9f));   // -> l_sl1
  }
  __syncthreads();

  if (tid < NEXP) {
    hist [blockIdx.x * NEXP + tid] = histL[tid];
    gsumB[blockIdx.x * NEXP + tid] = gsumL[tid];
  }
  if (tid < 4) scalB[blockIdx.x * 4 + tid] = scalL[tid];
}

// ---------------------------------------------------------------------------
// Kernel 3: per-expert exclusive scan over blocks + 135 stats + capacity_fp
// ---------------------------------------------------------------------------
__global__ __launch_bounds__(64) void k_scan(const int*   __restrict__ hist,
                                             const float* __restrict__ gsumB,
                                             const float* __restrict__ scalB,
                                             int*   __restrict__ offs,
                                             float* __restrict__ out,
                                             int out_size) {
  __shared__ float prodL[NEXP], g2L[NEXP];
  __shared__ int   cntL[NEXP], routedL[NEXP];
  __shared__ float scL[4];
  __shared__ float totRoutedInv;

  const int e = threadIdx.x;
  int run = 0; float gs = 0.f;
  for (int b = 0; b < NBLK; ++b) {
    offs[b * NEXP + e] = run;           // tokens of expert e in earlier blocks
    run += hist[b * NEXP + e];
    gs  += gsumB[b * NEXP + e];
  }
  const int routed = run < CAP ? run : CAP;   // capacity keeps the first 128
  cntL[e] = run; routedL[e] = routed;
  prodL[e] = gs * (float)run;                 // me*ce numerator
  g2L[e]   = gs * gs;                         // me^2 numerator
  if (e < 4) {
    float sc = 0.f;
    for (int b = 0; b < NBLK; ++b) sc += scalB[b * 4 + e];
    scL[e] = sc;
  }
  __syncthreads();

  if (e == 0) {
    float sp = 0.f, sg2 = 0.f; int tot = 0;
    for (int i = 0; i < NEXP; ++i) { sp += prodL[i]; sg2 += g2L[i]; tot += routedL[i]; }
    const float Sf = (float)S_TOK;
    const float c  = (float)NEXP / (Sf * Sf);
    out[0] = c * sp;           // l_aux = E * sum(me*ce)
    out[1] = scL[3] / Sf;      // l_sl1
    out[2] = c * sg2;          // l_mil = E * sum(me^2)
    out[3] = scL[0] / Sf;      // l_z
    out[4] = scL[1] / Sf;      // batch_entropy
    out[5] = scL[2] / Sf;      // batch_prob
    out[6] = (float)tot / Sf;  // fraction_routed
    totRoutedInv = 1.0f / (float)tot;
    out[out_size - 1] = 128.0f;  // capacity_fp = min(S, S/E*1.0) = 128.0
  }
  __syncthreads();

  out[7 + e]        = (float)cntL[e] / (float)S_TOK;       // expert_fraction
  out[7 + NEXP + e] = (float)routedL[e] * totRoutedInv;    // expert_fraction_routed
}

// ---------------------------------------------------------------------------
// Kernel 4: order-preserving rank within block + sparse scatter of outputs
// ---------------------------------------------------------------------------
__global__ __launch_bounds__(BLK_TOK) void k_scatter(const int*   __restrict__ idx,
                                                     const float* __restrict__ gmx,
                                                     const int*   __restrict__ offs,
                                                     float* __restrict__ out) {
  __shared__ int idxL[BLK_TOK];
  const int t = threadIdx.x;
  const int b = blockIdx.x;
  const int s = b * BLK_TOK + t;
  const int e = idx[s];
  idxL[t] = e;
  __syncthreads();

  int r = 0;
  for (int j = 0; j < t; ++j) r += (idxL[j] == e) ? 1 : 0;   // rank in s-order
  const int loc = offs[b * NEXP + e] + r;
  if (loc < CAP) {
    const float g = gmx[s];
    const size_t base = 135 + (((size_t)s * NEXP) + (size_t)e) * CAP + (size_t)loc;
    out[base] = g;                                    // combine_weights[s,e,loc]
    out[base + (size_t)S_TOK * NEXP * CAP] = 1.0f;    // dispatch_mask[s,e,loc]
  }
}

// ---------------------------------------------------------------------------
extern "C" void kernel_launch(void* const* d_in, const int* in_sizes, int n_in,
                              void* d_out, int out_size, void* d_ws, size_t ws_size,
                              hipStream_t stream) {
  const float* x = (const float*)d_in[0];   // [8192,1024]
  const float* W = (const float*)d_in[1];   // [1024,64]
  float* out = (float*)d_out;

  char* ws = (char*)d_ws;
  int*   idx   = (int*)  (ws + WS_IDX);
  float* gmx   = (float*)(ws + WS_GMX);
  int*   hist  = (int*)  (ws + WS_HIST);
  float* gsumB = (float*)(ws + WS_GSUM);
  float* scalB = (float*)(ws + WS_SCAL);
  int*   offs  = (int*)  (ws + WS_OFFS);

  const long n  = (long)out_size;
  const long n4 = n / 4;
  k_zero<<<4096, 256, 0, stream>>>((float4*)d_out, n4, out, n);
  k_gate<<<NBLK, 256, 0, stream>>>(x, W, idx, gmx, hist, gsumB, scalB);
  k_scan<<<1, 64, 0, stream>>>(hist, gsumB, scalB, offs, out, out_size);
  k_scatter<<<NBLK, BLK_TOK, 0, stream>>>(idx, gmx, offs, out);
}